// GraphModel_39230231281829
// MI455X (gfx1250) — compile-verified
//
#include <hip/hip_runtime.h>

typedef __attribute__((ext_vector_type(2))) float v2f;
typedef __attribute__((ext_vector_type(8))) float v8f;

#define NNODES 100000
#define DIN    256
#define DHID   97
#define LDP    112   // padded feature stride = 7 * 16

// ---------------------------------------------------------------------------
// deg[dst] += 1 over the edge list (float atomics; L2-resident)
__global__ void deg_count_kernel(const int* __restrict__ dst,
                                 float* __restrict__ deg, int ne) {
  int e = blockIdx.x * blockDim.x + threadIdx.x;
  if (e < ne) atomicAdd(&deg[dst[e]], 1.0f);
}

// dinv[i] = rsqrt(deg[i] + 1)   (+1 = self loop, so deg >= 1 always)
__global__ void dinv_kernel(float* __restrict__ deg, int n) {
  int i = blockIdx.x * blockDim.x + threadIdx.x;
  if (i < n) deg[i] = rsqrtf(deg[i] + 1.0f);
}

// Copy [rows x cols] weight into zero-padded [rowsP x ldp] buffer
__global__ void pad_w_kernel(const float* __restrict__ w, float* __restrict__ wp,
                             int rows, int cols, int rowsP, int ldp) {
  int idx = blockIdx.x * blockDim.x + threadIdx.x;
  if (idx >= rowsP * ldp) return;
  int r = idx / ldp, c = idx % ldp;
  wp[idx] = (r < rows && c < cols) ? w[r * cols + c] : 0.0f;
}

// ---------------------------------------------------------------------------
// fp32 WMMA GEMM: C[M x 112] = A[M x K] * B[K x 112]
// One wave computes one 16x16 tile. K must be a multiple of 4.
// Operand layout per CDNA5 ISA (16x4 f32 A, 4x16 f32 B, 16x16 f32 C/D):
//   A: lanes 0-15 hold M=lane, {v0,v1} = K{0,1}; lanes 16-31 = K{2,3}
//   B: lanes 0-15 hold N=lane, {v0,v1} = K{0,1}; lanes 16-31 = K{2,3}
//   C/D: vgpr r = row r (lanes 0-15) / row r+8 (lanes 16-31), N = lane&15
__global__ __launch_bounds__(128)
void gemm_wmma_f32_kernel(const float* __restrict__ A, int lda,
                          const float* __restrict__ B, int ldb,
                          float* __restrict__ C, int ldc,
                          int M, int K) {
  const int wave  = threadIdx.x >> 5;
  const int lane  = threadIdx.x & 31;
  const int mtile = blockIdx.x * (blockDim.x >> 5) + wave;
  const int ntile = blockIdx.y;
  if (mtile * 16 >= M) return;                 // wave-uniform: EXEC stays all-1s

  const int row   = mtile * 16 + (lane & 15);
  const int col   = ntile * 16 + (lane & 15);
  const int khalf = (lane >> 4) * 2;           // 0 for lanes 0-15, 2 for 16-31

  const float* __restrict__ arow = A + (size_t)row * lda;
  v8f acc = {};
  for (int k = 0; k < K; k += 4) {
    v2f a, b;
    a.x = arow[k + khalf];
    a.y = arow[k + khalf + 1];
    b.x = B[(size_t)(k + khalf)     * ldb + col];
    b.y = B[(size_t)(k + khalf + 1) * ldb + col];
    acc = __builtin_amdgcn_wmma_f32_16x16x4_f32(
        /*neg_a=*/false, a, /*neg_b=*/false, b,
        /*c_mod=*/(short)0, acc, /*reuse_a=*/false, /*reuse_b=*/false);
  }

  const int rbase = mtile * 16 + ((lane >> 4) * 8);
#pragma unroll
  for (int r = 0; r < 8; ++r)
    C[(size_t)(rbase + r) * ldc + col] = acc[r];
}

// ---------------------------------------------------------------------------
// Edge aggregation: agg[dst] += h[src] * dinv[src]*dinv[dst]
// One wave per edge; lanes stripe the 97 features. L2-resident atomics.
__global__ __launch_bounds__(256)
void aggregate_kernel(const float* __restrict__ h, int ldh,
                      float* __restrict__ agg, int ldo,
                      const int* __restrict__ src, const int* __restrict__ dst,
                      const float* __restrict__ dinv, int ne) {
  const int lane = threadIdx.x & 31;
  const int e = blockIdx.x * (blockDim.x >> 5) + (threadIdx.x >> 5);
  if (e >= ne) return;
  const int s = src[e], d = dst[e];
  const float w = dinv[s] * dinv[d];
  const float* __restrict__ hs = h + (size_t)s * ldh;
  float* __restrict__ ad = agg + (size_t)d * ldo;
  for (int f = lane; f < DHID; f += 32)
    atomicAdd(&ad[f], hs[f] * w);
}

// Self-loop contribution (runs after the atomic pass; plain RMW is safe
// because kernels on the same stream serialize). Optionally folds in bias.
__global__ __launch_bounds__(128)
void self_loop_kernel(const float* __restrict__ h, int ldh,
                      float* __restrict__ agg, int ldo,
                      const float* __restrict__ dinv,
                      const float* __restrict__ bias) {
  const int i = blockIdx.x;
  const int f = threadIdx.x;
  if (f >= DHID) return;
  float w = dinv[i];
  w *= w;
  float v = agg[(size_t)i * ldo + f] + h[(size_t)i * ldh + f] * w;
  if (bias) v += bias[f];
  agg[(size_t)i * ldo + f] = v;
}

// h = relu(h + b) in place (padded cols 97..111 stay zero from memset)
__global__ __launch_bounds__(128)
void bias_relu_kernel(float* __restrict__ h, int ld, const float* __restrict__ b) {
  const int i = blockIdx.x;
  const int f = threadIdx.x;
  if (f >= DHID) return;
  float v = h[(size_t)i * ld + f] + b[f];
  h[(size_t)i * ld + f] = v > 0.0f ? v : 0.0f;
}

// ---------------------------------------------------------------------------
extern "C" void kernel_launch(void* const* d_in, const int* in_sizes, int n_in,
                              void* d_out, int out_size, void* d_ws, size_t ws_size,
                              hipStream_t stream) {
  const float* x     = (const float*)d_in[0];
  const int*   edges = (const int*)d_in[1];
  const float* W1    = (const float*)d_in[2];
  const float* b1    = (const float*)d_in[3];
  const float* W2    = (const float*)d_in[4];
  const float* b2    = (const float*)d_in[5];
  float*       out   = (float*)d_out;

  const int ne = in_sizes[1] / 2;            // 3.2M
  const int* src = edges;                    // edge_index[0]
  const int* dst = edges + ne;               // edge_index[1]

  // Workspace layout (float offsets, 256-float aligned)
  float* ws = (float*)d_ws;
  const size_t OFF_DINV = 0;                        // 100000
  const size_t OFF_W1P  = 100096;                   // 256*112 = 28672
  const size_t OFF_W2P  = OFF_W1P + 28672;          // 112*112 = 12544
  const size_t OFF_BUFA = OFF_W2P + 12544;          // 100000*112
  const size_t OFF_BUFB = OFF_BUFA + (size_t)NNODES * LDP;
  float* dinv = ws + OFF_DINV;
  float* W1p  = ws + OFF_W1P;
  float* W2p  = ws + OFF_W2P;
  float* bufA = ws + OFF_BUFA;   // h1, later t2
  float* bufB = ws + OFF_BUFB;   // agg1 -> h2

  // --- normalization coefficients ---
  hipMemsetAsync(dinv, 0, (size_t)NNODES * sizeof(float), stream);
  deg_count_kernel<<<(ne + 255) / 256, 256, 0, stream>>>(dst, dinv, ne);
  dinv_kernel<<<(NNODES + 255) / 256, 256, 0, stream>>>(dinv, NNODES);

  // --- pad weights into WMMA-friendly shapes ---
  pad_w_kernel<<<(DIN * LDP + 255) / 256, 256, 0, stream>>>(W1, W1p, DIN, DHID, DIN, LDP);
  pad_w_kernel<<<(LDP * LDP + 255) / 256, 256, 0, stream>>>(W2, W2p, DHID, DHID, LDP, LDP);

  const int mtiles = NNODES / 16;            // 6250 exactly
  dim3 ggrid((mtiles + 3) / 4, LDP / 16);    // 4 waves (M-tiles) per block, 7 N-tiles

  // --- layer 1: h1 = x @ W1 ---
  gemm_wmma_f32_kernel<<<ggrid, 128, 0, stream>>>(x, DIN, W1p, LDP, bufA, LDP, NNODES, DIN);

  // --- layer 1 aggregation into bufB ---
  hipMemsetAsync(bufB, 0, (size_t)NNODES * LDP * sizeof(float), stream);
  aggregate_kernel<<<(ne + 7) / 8, 256, 0, stream>>>(bufA, LDP, bufB, LDP, src, dst, dinv, ne);
  self_loop_kernel<<<NNODES, 128, 0, stream>>>(bufA, LDP, bufB, LDP, dinv, nullptr);
  bias_relu_kernel<<<NNODES, 128, 0, stream>>>(bufB, LDP, b1);

  // --- layer 2: t2 = h2 @ W2 (K padded to 112; pad cols/rows are zero) ---
  gemm_wmma_f32_kernel<<<ggrid, 128, 0, stream>>>(bufB, LDP, W2p, LDP, bufA, LDP, NNODES, LDP);

  // --- layer 2 aggregation directly into d_out (ld = 97) ---
  hipMemsetAsync(out, 0, (size_t)NNODES * DHID * sizeof(float), stream);
  aggregate_kernel<<<(ne + 7) / 8, 256, 0, stream>>>(bufA, LDP, out, DHID, src, dst, dinv, ne);
  self_loop_kernel<<<NNODES, 128, 0, stream>>>(bufA, LDP, out, DHID, dinv, b2);
}